// SignatureLayer_48473000902869
// MI455X (gfx1250) — compile-verified
//
#include <hip/hip_runtime.h>
#include <hip/hip_bf16.h>

// Path-signature (depth 3) of 1024 paths, L=2048, 7 channels + time channel.
// Memory-bound (58.7 MB in; fits in 192 MB L2). sig2 = Excl^T @ Inc is a
// (8 x T) @ (T x 8) K-reduction GEMM mapped onto V_WMMA_F32_16X16X4_F32.
// Each wave packs TWO batches into one 16x16 tile (rows/cols 0-7 = b0,
// 8-15 = b1); the diagonal 8x8 blocks of D are the per-batch sig2 partials.
// Hot loop: clamp-free pointer-increment addressing (immediate offsets),
// branchless time-channel select, WMMA accumulate. Only the w==7 wave runs a
// 4-step clamped tail for the chunk-15 boundary (wave-uniform branch).

typedef __attribute__((ext_vector_type(2))) float v2f;
typedef __attribute__((ext_vector_type(8))) float v8f;

#define SIG_B      1024   // batches
#define SIG_L      2048   // path length
#define SIG_C      7      // input channels (d = 8 with time)
#define SIG_T      2047   // number of increments
#define SIG_HC     128    // increments per chunk
#define SIG_NH     16     // chunks per batch
#define SIG_NWAVE  8      // waves per batch-pair; each wave owns 2 chunks
#define SIG_COLS   136    // 8 + 64 + 64 outputs per batch
#define SIG_DT     (1.0f / 2047.0f)

__device__ __forceinline__ void sig_step2(float& S, float bA, float bB, v8f& acc)
{
    v2f a, bv;
    a.x = S; bv.x = bA; S += bA;
    a.y = S; bv.y = bB; S += bB;
    acc = __builtin_amdgcn_wmma_f32_16x16x4_f32(
              false, a, false, bv, (short)0, acc, false, false);
}

// ---------------------------------------------------------------------------
// Stage A: one wave per (batch-pair, chunk-pair). Lane = h*16 + q:
//   h = lane>>4   -> chunk 2w+h   (K-slots 2h,2h+1 of each WMMA)
//   p = q>>3      -> batch 2bb+p  (tile rows/cols p*8 .. p*8+7)
//   cm = q&7      -> component (0 = synthesized time, 1..7 = path channel cm-1)
// ---------------------------------------------------------------------------
__global__ __launch_bounds__(32)
void sig_partial_kernel(const float* __restrict__ path,
                        float* __restrict__ ws_sums,   // [B][NH][8] chunk sums
                        float* __restrict__ ws_sig2)   // [B][NWAVE][64] wmma partials
{
    const int bb   = blockIdx.x >> 3;      // batch pair index (0..511)
    const int w    = blockIdx.x & 7;
    const int lane = threadIdx.x;
    const int q    = lane & 15;
    const int p    = q >> 3;               // batch sub-index within pair
    const int cm   = q & 7;                // component 0..7

    const int batch = 2 * bb + p;
    const int chunk = 2 * w + (lane >> 4);
    const int t0    = chunk * SIG_HC;
    const bool is_time = (cm == 0);

    const int comp = (cm >= 1) ? (cm - 1) : 0;   // time/dup lanes alias ch 0
    const float* __restrict__ bbase = path + (size_t)batch * SIG_L * SIG_C;

    const float* __restrict__ ptr = bbase + (t0 + 1) * SIG_C + comp; // x[t0+1][comp]

    float prev = bbase[t0 * SIG_C + comp];         // x[t0][comp]
    float S = 0.0f;                                // local exclusive prefix
    v8f acc = (v8f)0.0f;

    // main loop: no clamping needed (max row read = x[t0 + 4*iters] <= x[2047])
    const int iters = (w == 7) ? 31 : 32;          // 4 timesteps per iteration
    for (int i = 0; i < iters; ++i) {
        const float x0 = ptr[0];
        const float x1 = ptr[SIG_C];
        const float x2 = ptr[2 * SIG_C];
        const float x3 = ptr[3 * SIG_C];
        ptr += 4 * SIG_C;

        const float b0 = is_time ? SIG_DT : (x0 - prev);
        const float b1 = is_time ? SIG_DT : (x1 - x0);
        const float b2 = is_time ? SIG_DT : (x2 - x1);
        const float b3 = is_time ? SIG_DT : (x3 - x2);
        prev = x3;

        sig_step2(S, b0, b1, acc);
        sig_step2(S, b2, b3, acc);
    }

    // tail (wave-uniform): steps 124..127 of chunks 14/15 with address clamp.
    // Chunk 14 clamp is a no-op; chunk 15 step 127 reads x[2047] twice so its
    // path increments are exactly 0 (time lane's spurious dt fixed in stage B).
    if (w == 7) {
        int off = (t0 + 125) * SIG_C + comp;
        const int offMax = (SIG_L - 1) * SIG_C + comp;
        const int o0 = (off             < offMax) ? off             : offMax;
        const int o1 = (off +   SIG_C   < offMax) ? off +   SIG_C   : offMax;
        const int o2 = (off + 2*SIG_C   < offMax) ? off + 2*SIG_C   : offMax;
        const int o3 = (off + 3*SIG_C   < offMax) ? off + 3*SIG_C   : offMax;
        const float x0 = bbase[o0];
        const float x1 = bbase[o1];
        const float x2 = bbase[o2];
        const float x3 = bbase[o3];

        const float b0 = is_time ? SIG_DT : (x0 - prev);
        const float b1 = is_time ? SIG_DT : (x1 - x0);
        const float b2 = is_time ? SIG_DT : (x2 - x1);
        const float b3 = is_time ? SIG_DT : (x3 - x2);

        sig_step2(S, b0, b1, acc);
        sig_step2(S, b2, b3, acc);
    }

    // chunk sums: every lane holds a valid (batch, chunk, component) sum
    ws_sums[((size_t)batch * SIG_NH + chunk) * 8 + cm] = S;

    // D layout (32-bit C/D): lanes 0-15: VGPR j -> D[j][lane];
    //                        lanes 16-31: VGPR j -> D[8+j][lane-16].
    // b0 partial = D[0..7][0..7]   -> lanes 0..7
    // b1 partial = D[8..15][8..15] -> lanes 24..31
    if (lane < 8) {
        float* dst = ws_sig2 + ((size_t)(2 * bb) * SIG_NWAVE + w) * 64;
        #pragma unroll
        for (int j = 0; j < 8; ++j)
            dst[j * 8 + lane] = acc[j];
    } else if (lane >= 24) {
        float* dst = ws_sig2 + ((size_t)(2 * bb + 1) * SIG_NWAVE + w) * 64;
        #pragma unroll
        for (int j = 0; j < 8; ++j)
            dst[j * 8 + (lane - 24)] = acc[j];
    }
}

// ---------------------------------------------------------------------------
// Stage B: one block per batch. Fix the analytically-known spurious step of
// chunk 15 (time channel ran 128 steps instead of 127), scan the 16 chunk
// sums, add cross terms P_c (x) s_c, sum the 8 WMMA partials, emit outputs.
// ---------------------------------------------------------------------------
__global__ __launch_bounds__(64)
void sig_combine_kernel(const float* __restrict__ ws_sums,
                        const float* __restrict__ ws_sig2,
                        float* __restrict__ out, int out_size)
{
    const int b   = blockIdx.x;
    const int tid = threadIdx.x;          // 64 threads: tid = d*8 + e
    __shared__ float s[SIG_NH * 8];       // 128 floats

    const float* bs = ws_sums + (size_t)b * SIG_NH * 8;
    s[tid] = bs[tid];
    {
        float v = bs[tid + 64];
        if (tid + 64 == 15 * 8 + 0) v -= SIG_DT;   // chunk 15 time sum: 128dt -> 127dt
        s[tid + 64] = v;
    }
    __syncthreads();

    const int d = tid >> 3;
    const int e = tid & 7;

    float P = 0.0f;       // prefix of chunk sums, component d
    float cross = 0.0f;   // sum_c P_c[d] * s_c[e]
    #pragma unroll
    for (int c = 0; c < SIG_NH; ++c) {
        cross += P * s[c * 8 + e];
        P     += s[c * 8 + d];
    }
    // P is now sig1[d]

    float part = 0.0f;
    const float* bp = ws_sig2 + (size_t)b * SIG_NWAVE * 64;
    #pragma unroll
    for (int w = 0; w < SIG_NWAVE; ++w)
        part += bp[w * 64 + tid];

    // spurious WMMA step (chunk 15, t=2047): added s15[d]*dt into column e=0
    if (e == 0) part -= s[15 * 8 + d] * SIG_DT;

    const float sig2_de = part + cross;
    const size_t base = (size_t)b * SIG_COLS;

    if (base + 8 + tid < (size_t)out_size)
        out[base + 8 + tid] = sig2_de;                 // sig2, row-major (d,e)
    if (base + 72 + tid < (size_t)out_size)
        out[base + 72 + tid] = P * sig2_de;            // sig3[d,e] = sig1[d]*sig2[d,e]
    if (e == 0 && base + d < (size_t)out_size)
        out[base + d] = P;                             // sig1[d]
}

extern "C" void kernel_launch(void* const* d_in, const int* in_sizes, int n_in,
                              void* d_out, int out_size, void* d_ws, size_t ws_size,
                              hipStream_t stream) {
    (void)in_sizes; (void)n_in; (void)ws_size;
    const float* path = (const float*)d_in[0];

    float* ws_sums = (float*)d_ws;                         // [1024][16][8]
    float* ws_sig2 = ws_sums + (size_t)SIG_B * SIG_NH * 8; // [1024][8][64]

    // one wave per (batch pair, chunk pair): 512 * 8 = 4096 waves
    sig_partial_kernel<<<dim3((SIG_B / 2) * SIG_NWAVE), dim3(32), 0, stream>>>(
        path, ws_sums, ws_sig2);
    sig_combine_kernel<<<dim3(SIG_B), dim3(64), 0, stream>>>(
        ws_sums, ws_sig2, (float*)d_out, out_size);
}